// AllZeroDigitalFilter_5695126634547
// MI455X (gfx1250) — compile-verified
//
#include <hip/hip_runtime.h>

// AllZeroDigitalFilter (time-varying FIR) for MI455X / gfx1250.
//   y[b,t] = sum_{m=0..49} x[b, t-m] * h[b,t,m]
//   h[b,t,m] = (1-w)*coef[b,n,m] + w*coef[b,min(n+1,N-1),m],
//   n = t/80, w = (t mod 80)/80.
//
// One wave32 computes 16 consecutive outputs via V_WMMA_F32_16X16X4_F32.
// K axis reversed so A is ascending in memory (contiguous b64 loads):
//   A[i][k] = x[t0 + i - 51 + k]          (k = 0..51)
//   B[k][0] = coefL[51-k], B[k][1] = coefR[51-k]   (zero where 51-k > 49)
//   C[:,0]/C[:,1] = dots vs left/right frame coeffs -> blended by w.
// All 13 A pairs + 13 B pairs are preloaded into registers, then 13
// back-to-back WMMAs accumulate; boundary (t0<51) blocks take a clamped,
// branch-free path.

typedef __attribute__((ext_vector_type(2))) float v2f;
typedef __attribute__((ext_vector_type(2), aligned(4))) float v2f_u;  // 4B-aligned b64 loads
typedef __attribute__((ext_vector_type(8))) float v8f;

#define FILT_D   50                   // FILTER_ORDER + 1
#define PERIOD   80                   // FRAME_PERIOD
#define NFRAMES  3000
#define TLEN     (NFRAMES * PERIOD)   // 240000
#define KPAD     52                   // 50 taps padded to multiple of 4
#define NIT      (KPAD / 4)           // 13 WMMA steps

__global__ __launch_bounds__(256)
void azdf_wmma_kernel(const float* __restrict__ x,
                      const float* __restrict__ coef,
                      float* __restrict__ out)
{
    __shared__ float sc[8][32];   // per-wave unpack: [rows 0..15 col0 | rows 0..15 col1]

    const int lane  = threadIdx.x & 31;
    const int warp  = threadIdx.x >> 5;   // 8 waves per block = 8 batches
    const int batch = warp;
    const int t0    = blockIdx.x * 16;    // 15000 blocks cover T=240000
    const int row   = lane & 15;          // A-matrix M index / C-matrix N index
    const int col   = row;                // B/C column this lane covers
    const int khi   = (lane >> 4) << 1;   // lanes 16..31 carry K = kbase+2, kbase+3

    const int n  = t0 / PERIOD;                                   // uniform per block
    const int nR = (n + 1 < NFRAMES) ? (n + 1) : (NFRAMES - 1);   // replicate-pad last frame

    const float* __restrict__ xb = x    + batch * TLEN;
    const float* __restrict__ bL = coef + (batch * NFRAMES + n ) * FILT_D;
    const float* __restrict__ bR = coef + (batch * NFRAMES + nR) * FILT_D;

    const int base = t0 + row - (KPAD - 1);   // x index of A[row][k=0]

    // ---- Preload all A operand pairs (ascending k -> contiguous memory) ----
    v2f aArr[NIT];
    if (t0 >= KPAD - 1) {
        // Interior: every index >= t0 - 51 >= 0; straight b64 loads.
#pragma unroll
        for (int it = 0; it < NIT; ++it) {
            const int k0 = 4 * it + khi;
            aArr[it] = *(const v2f_u*)(xb + base + k0);
        }
    } else {
        // Boundary blocks (4 of 15000): clamp index so the load is always
        // valid, zero the value with a cndmask. No exec-mask diamonds.
#pragma unroll
        for (int it = 0; it < NIT; ++it) {
            const int k0 = 4 * it + khi;
            const int i0 = base + k0;
            const int i1 = i0 + 1;
            float ax = xb[i0 >= 0 ? i0 : 0];
            float ay = xb[i1 >= 0 ? i1 : 0];
            aArr[it].x = (i0 >= 0) ? ax : 0.0f;
            aArr[it].y = (i1 >= 0) ? ay : 0.0f;
        }
    }

    // ---- Preload all B operand pairs (only columns 0 and 1 are non-zero) ----
    v2f bArr[NIT];
#pragma unroll
    for (int it = 0; it < NIT; ++it) { bArr[it].x = 0.0f; bArr[it].y = 0.0f; }
    if (col < 2) {                       // single branch, executed by 4 lanes
        const float* __restrict__ bp = (col == 0) ? bL : bR;
#pragma unroll
        for (int it = 0; it < NIT; ++it) {
            const int k0 = 4 * it + khi;
            const int k1 = k0 + 1;
            // coef index m = 51 - k; valid taps are m <= 49 (i.e. k >= 2);
            // addresses 51-k >= 0 are always safe, so mask with cndmask only.
            float c0 = bp[(KPAD - 1) - k0];
            float c1 = bp[(KPAD - 1) - k1];
            bArr[it].x = (k0 >= 2) ? c0 : 0.0f;
            bArr[it].y = (k1 >= 2) ? c1 : 0.0f;
        }
    }

    // ---- 13 back-to-back fp32 WMMAs: D = A x B + C ----
    v8f c = {};
#pragma unroll
    for (int it = 0; it < NIT; ++it) {
        c = __builtin_amdgcn_wmma_f32_16x16x4_f32(
                false, aArr[it], false, bArr[it], (short)0, c, false, false);
    }

    // C/D f32 layout: VGPR j, lanes 0..15 -> (M=j,   N=lane),
    //                 VGPR j, lanes 16..31 -> (M=8+j, N=lane-16).
    // Useful columns: N=0 (lanes 0,16) and N=1 (lanes 1,17).
    if (col < 2) {
        const int rbase = (lane >= 16) ? 8 : 0;
#pragma unroll
        for (int j = 0; j < 8; ++j)
            sc[warp][col * 16 + rbase + j] = c[j];
    }
    __syncthreads();

    if (lane < 16) {
        const float y0 = sc[warp][lane];        // dot(x window, coef[n])
        const float y1 = sc[warp][16 + lane];   // dot(x window, coef[nR])
        const float w  = (float)(t0 - n * PERIOD + lane) * (1.0f / (float)PERIOD);
        out[batch * TLEN + t0 + lane] = (1.0f - w) * y0 + w * y1;
    }
}

extern "C" void kernel_launch(void* const* d_in, const int* in_sizes, int n_in,
                              void* d_out, int out_size, void* d_ws, size_t ws_size,
                              hipStream_t stream)
{
    const float* x    = (const float*)d_in[0];   // (8, 240000) f32
    const float* coef = (const float*)d_in[1];   // (8, 3000, 50) f32
    float* out        = (float*)d_out;           // (8, 240000) f32

    const int blocks = TLEN / 16;                // 15000; wave w in block = batch w
    azdf_wmma_kernel<<<blocks, 256, 0, stream>>>(x, coef, out);
}